// CollisionlessEmbedding_15745350107436
// MI455X (gfx1250) — compile-verified
//
#include <hip/hip_runtime.h>
#include <stdint.h>

// Hashed-embedding gather for MI455X (gfx1250, wave32).
//
// Layout insight: one output row = 128 floats = 512 B = 32 lanes x 16 B.
//   lane = half*16 + sub  (half selects table0/table1, sub selects 16B chunk)
//   output offset for lane = row*128 + lane*4  -> one coalesced 512B wave store.
// Each gather read is a contiguous 256 B table sub-row covered by 16 lanes of
// global_load_b128 -- optimal granularity for random HBM access.
//
// Purely bandwidth-bound (~840 MB total traffic, ~36 us floor @ 23.3 TB/s).
// gfx1250-specific paths used: global_load_b128/global_store_b128 (wave32
// shaped), global_prefetch_b8 (hash next tile's ids early, prefetch random
// gather targets toward L2), NT store hint for the streaming output.

typedef __attribute__((ext_vector_type(4))) float v4f;
typedef __attribute__((ext_vector_type(4))) int   v4i;

static constexpr uint32_t kNumEmb = 1000000u;
static constexpr uint32_t kSeed   = 42u;
static constexpr int kSubDim = 64;    // floats per table row
static constexpr int kEmbDim = 128;   // floats per output row
static constexpr int kRows   = 4;     // rows per wave per iteration (MLP)

__device__ __forceinline__ uint32_t hash_id(uint32_t x, uint32_t seed) {
  x ^= seed;
  x = (x ^ (x >> 16)) * 0x7FEB352Du;
  x = (x ^ (x >> 15)) * 0x846CA68Bu;
  x ^= x >> 16;
  return x % kNumEmb;
}

__global__ __launch_bounds__(256) void emb_gather_kernel(
    const int* __restrict__ ids, const float* __restrict__ t0,
    const float* __restrict__ t1, float* __restrict__ out, int n_rows)
{
  const int lane = (int)(threadIdx.x & 31u);
  const int wave = (int)(threadIdx.x >> 5u);
  const int wpb  = (int)(blockDim.x >> 5u);
  const int wid  = (int)blockIdx.x * wpb + wave;
  const int nw   = (int)gridDim.x * wpb;

  const uint32_t half = (uint32_t)(lane >> 4);   // 0 -> table0, 1 -> table1
  const uint32_t sub  = (uint32_t)(lane & 15);   // 16B chunk within 256B sub-row
  const float* __restrict__ table = half ? t1 : t0;
  const uint32_t seed = kSeed + half;
  const uint32_t col  = sub * 4u;                // float offset within sub-row

  const int stride = nw * kRows;

  for (int base = wid * kRows; base < n_rows; base += stride) {
    if (base + kRows <= n_rows) {
      // ids for current tile: one 16B load (all lanes broadcast-coalesce).
      v4i id4 = *(const v4i*)(ids + base);

      // Prefetch next tile's random gather targets toward L2.
      // Hash recompute is ~8 VALU ops; prefetch is fire-and-forget
      // (global_prefetch_b8, no LOADcnt).
      const int nbase = base + stride;
      if (nbase + kRows <= n_rows) {
        v4i nid4 = *(const v4i*)(ids + nbase);
#pragma unroll
        for (int k = 0; k < kRows; ++k) {
          uint32_t idx = hash_id((uint32_t)nid4[k], seed);
          __builtin_prefetch(table + (size_t)idx * kSubDim + col, 0, 1);
        }
      }

      // 4 independent 128-bit gathers in flight before any store waits.
      v4f v[kRows];
#pragma unroll
      for (int k = 0; k < kRows; ++k) {
        uint32_t idx = hash_id((uint32_t)id4[k], seed);
        v[k] = *(const v4f*)(table + (size_t)idx * kSubDim + col);
      }

      // Coalesced 512B-per-row NT stores (streaming output; keep L2 for
      // gather lines / prefetches).
#pragma unroll
      for (int k = 0; k < kRows; ++k) {
        __builtin_nontemporal_store(
            v[k], (v4f*)(out + (size_t)(base + k) * kEmbDim + (size_t)lane * 4u));
      }
    } else {
      // Tail: per-row scalar path with bounds checks.
      for (int r = base; r < n_rows; ++r) {
        uint32_t idx = hash_id((uint32_t)ids[r], seed);
        v4f v = *(const v4f*)(table + (size_t)idx * kSubDim + col);
        __builtin_nontemporal_store(
            v, (v4f*)(out + (size_t)r * kEmbDim + (size_t)lane * 4u));
      }
    }
  }
}

extern "C" void kernel_launch(void* const* d_in, const int* in_sizes, int n_in,
                              void* d_out, int out_size, void* d_ws, size_t ws_size,
                              hipStream_t stream) {
  const int*   ids = (const int*)d_in[0];
  const float* t0  = (const float*)d_in[1];
  const float* t1  = (const float*)d_in[2];
  float* out = (float*)d_out;
  const int n_rows = in_sizes[0];   // 4096*200 = 819200 flat ids

  const int threads = 256;                                  // 8 wave32 per block
  const int rows_per_block_iter = (threads / 32) * kRows;   // 32 rows/block/iter
  int blocks = (n_rows + rows_per_block_iter - 1) / rows_per_block_iter;
  if (blocks > 2048) blocks = 2048;  // grid-stride: ~12.5 iters/wave at N=819200
  if (blocks < 1) blocks = 1;

  emb_gather_kernel<<<blocks, threads, 0, stream>>>(ids, t0, t1, out, n_rows);
}